// HGTEncoder_90168543412783
// MI455X (gfx1250) — compile-verified
//
#include <hip/hip_runtime.h>
#include <hip/hip_bf16.h>
#include <math.h>

typedef __attribute__((ext_vector_type(2))) float v2f;
typedef __attribute__((ext_vector_type(8))) float v8f;

#if defined(__gfx1250__) && __has_builtin(__builtin_amdgcn_wmma_f32_16x16x4_f32)
#define HAVE_WMMA_F32 1
#else
#define HAVE_WMMA_F32 0
#endif

#define KC 16          // K-chunk staged in LDS per block iteration
#define LDSW (KC + 2)  // padded k-stride: 16-lane b64 reads are conflict-free

// ---------------------------------------------------------------------------
// GEMM + bias:  C[M,N] = A[M,K] @ W[K,N] + bias[N]
// Requirements: M % 32 == 0, N % 64 == 0, K % KC == 0.
// Block = 128 threads (4 waves); block tile = 128 rows x 64 cols.
// Wave w computes rows [bx*128 + w*32, +32) (two 16-row tiles) x 64 cols
// (4 col tiles) => 8 WMMA per K-step for 2 global b64 (A) + 4 ds b64 (B).
// W chunk (KC x 64) is staged TRANSPOSED in LDS: ws[c*LDSW + k], so each B
// fragment is a single aligned ds_load_b64 (no VGPR re-pairing movs).
// Global staging loads are coalesced float2 (params are only 8B-aligned due
// to odd-sized skip scalars in the flat param buffer).
// Tail waves (row0 >= M) still run staging + barriers; compute region is a
// wave-uniform branch so EXEC is all-ones inside WMMA.
// ---------------------------------------------------------------------------
__global__ void hgt_gemm_bias(const float* __restrict__ A,
                              const float* __restrict__ W,
                              const float* __restrict__ bias,
                              float* __restrict__ C,
                              int M, int K, int N) {
  __shared__ float ws[64 * LDSW];
  const int tid = threadIdx.x;
  const int lane = tid & 31;
  const int wave = tid >> 5;
  const int row0 = blockIdx.x * 128 + wave * 32;
  const int col0 = blockIdx.y * 64;
  const bool active = row0 < M;          // wave-uniform; M % 32 == 0
  const int nlo = lane & 15;
  const int rbase = row0 + ((lane >> 4) << 3);
#if HAVE_WMMA_F32
  const int m0 = row0 + nlo;
  const int koff = (lane >> 4) * 2;      // A/B fragment K: {0,1} or {2,3}
  const float* Arow0 = A + (size_t)(active ? m0 : 0) * K;
  const float* Arow1 = A + (size_t)(active ? (m0 + 16) : 0) * K;
  v8f acc[2][4] = {};
#else
  float acc[2][4][8] = {};
#endif
  for (int k0 = 0; k0 < K; k0 += KC) {
    // ---- stage W[k0..k0+KC) x [col0..col0+64) transposed into LDS ----
#pragma unroll
    for (int i = 0; i < (KC * 64) / (128 * 2); ++i) {  // 4 iters, float2 each
      int f = (i * 128 + tid) * 2;
      int r = f >> 6, c = f & 63;
      float2 wv =
          *reinterpret_cast<const float2*>(&W[(size_t)(k0 + r) * N + col0 + c]);
      ws[(c + 0) * LDSW + r] = wv.x;
      ws[(c + 1) * LDSW + r] = wv.y;
    }
    __syncthreads();
    if (active) {
#if HAVE_WMMA_F32
#pragma unroll
      for (int k = 0; k < KC; k += 4) {
        v2f a0 = *reinterpret_cast<const v2f*>(Arow0 + k0 + k + koff);
        v2f a1 = *reinterpret_cast<const v2f*>(Arow1 + k0 + k + koff);
#pragma unroll
        for (int j = 0; j < 4; ++j) {
          v2f b = *reinterpret_cast<const v2f*>(
              &ws[(j * 16 + nlo) * LDSW + k + koff]);
          acc[0][j] = __builtin_amdgcn_wmma_f32_16x16x4_f32(
              false, a0, false, b, (short)0, acc[0][j], false, false);
          acc[1][j] = __builtin_amdgcn_wmma_f32_16x16x4_f32(
              false, a1, false, b, (short)0, acc[1][j], false, false);
        }
      }
#else
      for (int k = 0; k < KC; ++k) {
#pragma unroll
        for (int j = 0; j < 4; ++j) {
          float wv = ws[(j * 16 + nlo) * LDSW + k];
#pragma unroll
          for (int r = 0; r < 8; ++r) {
            acc[0][j][r] += A[(size_t)(rbase + r) * K + k0 + k] * wv;
            acc[1][j][r] += A[(size_t)(rbase + 16 + r) * K + k0 + k] * wv;
          }
        }
      }
#endif
    }
    __syncthreads();
  }
  if (active) {
#pragma unroll
    for (int j = 0; j < 4; ++j) {
      float bv = bias ? bias[col0 + j * 16 + nlo] : 0.0f;
#pragma unroll
      for (int r = 0; r < 8; ++r) {
        C[(size_t)(rbase + r) * N + col0 + j * 16 + nlo] = acc[0][j][r] + bv;
        C[(size_t)(rbase + 16 + r) * N + col0 + j * 16 + nlo] = acc[1][j][r] + bv;
      }
    }
  }
}

// ---------------------------------------------------------------------------
// Head-blocked GEMM: out[n, h*64+e] = sum_d A[n, h*64+d] * R[h][d][e]
// A,out: [M,256] (4 heads x 64), R: [4][64][64]. Block tile = 128 rows x one
// head's 64 columns (grid.y = head). Same transposed-LDS scheme, K = 64.
// ---------------------------------------------------------------------------
__global__ void hgt_head_gemm(const float* __restrict__ A,
                              const float* __restrict__ R,
                              float* __restrict__ Cout, int M) {
  __shared__ float ws[64 * LDSW];
  const int tid = threadIdx.x;
  const int lane = tid & 31;
  const int wave = tid >> 5;
  const int row0 = blockIdx.x * 128 + wave * 32;
  const int h = blockIdx.y;
  const bool active = row0 < M;          // wave-uniform; M % 32 == 0
  const int nlo = lane & 15;
  const int rbase = row0 + ((lane >> 4) << 3);
  const float* Rh = R + (size_t)h * 64 * 64;
#if HAVE_WMMA_F32
  const int m0 = row0 + nlo;
  const int koff = (lane >> 4) * 2;
  const float* Arow0 = A + (size_t)(active ? m0 : 0) * 256 + h * 64;
  const float* Arow1 = A + (size_t)(active ? (m0 + 16) : 0) * 256 + h * 64;
  v8f acc[2][4] = {};
#else
  float acc[2][4][8] = {};
#endif
  for (int d0 = 0; d0 < 64; d0 += KC) {
#pragma unroll
    for (int i = 0; i < (KC * 64) / (128 * 2); ++i) {
      int f = (i * 128 + tid) * 2;
      int r = f >> 6, c = f & 63;
      float2 wv =
          *reinterpret_cast<const float2*>(&Rh[(size_t)(d0 + r) * 64 + c]);
      ws[(c + 0) * LDSW + r] = wv.x;
      ws[(c + 1) * LDSW + r] = wv.y;
    }
    __syncthreads();
    if (active) {
#if HAVE_WMMA_F32
#pragma unroll
      for (int k = 0; k < KC; k += 4) {
        v2f a0 = *reinterpret_cast<const v2f*>(Arow0 + d0 + k + koff);
        v2f a1 = *reinterpret_cast<const v2f*>(Arow1 + d0 + k + koff);
#pragma unroll
        for (int j = 0; j < 4; ++j) {
          v2f b = *reinterpret_cast<const v2f*>(
              &ws[(j * 16 + nlo) * LDSW + k + koff]);
          acc[0][j] = __builtin_amdgcn_wmma_f32_16x16x4_f32(
              false, a0, false, b, (short)0, acc[0][j], false, false);
          acc[1][j] = __builtin_amdgcn_wmma_f32_16x16x4_f32(
              false, a1, false, b, (short)0, acc[1][j], false, false);
        }
      }
#else
      for (int k = 0; k < KC; ++k) {
#pragma unroll
        for (int j = 0; j < 4; ++j) {
          float rv = ws[(j * 16 + nlo) * LDSW + k];
#pragma unroll
          for (int r = 0; r < 8; ++r) {
            acc[0][j][r] += A[(size_t)(rbase + r) * 256 + h * 64 + d0 + k] * rv;
            acc[1][j][r] +=
                A[(size_t)(rbase + 16 + r) * 256 + h * 64 + d0 + k] * rv;
          }
        }
      }
#endif
    }
    __syncthreads();
  }
  if (active) {
#pragma unroll
    for (int j = 0; j < 4; ++j)
#pragma unroll
      for (int r = 0; r < 8; ++r) {
        Cout[(size_t)(rbase + r) * 256 + h * 64 + j * 16 + nlo] = acc[0][j][r];
        Cout[(size_t)(rbase + 16 + r) * 256 + h * 64 + j * 16 + nlo] =
            acc[1][j][r];
      }
  }
}

// ---------------------------------------------------------------------------
// Edge pass A: alpha[e,h] = <k_rel[src,h,:], q[dst,h,:]> * prior[h] * scale
// and running ordered-uint atomicMax of alpha per (dst,h).
// ---------------------------------------------------------------------------
__device__ __forceinline__ unsigned f32_ordered_key(float f) {
  unsigned u = __float_as_uint(f);
  return (u & 0x80000000u) ? ~u : (u | 0x80000000u);
}
__device__ __forceinline__ float f32_ordered_unkey(unsigned key) {
  unsigned u = (key & 0x80000000u) ? (key & 0x7fffffffu) : ~key;
  return __uint_as_float(u);
}

__global__ void hgt_alpha_kernel(const float* __restrict__ krel,
                                 const float* __restrict__ q,
                                 const int* __restrict__ src,
                                 const int* __restrict__ dst,
                                 const float* __restrict__ prior,
                                 float scale,
                                 float* __restrict__ alphabuf,
                                 unsigned* __restrict__ amaxU,
                                 int E) {
  int t = blockIdx.x * blockDim.x + threadIdx.x;
  if (t >= E * 4) return;
  int e = t >> 2, h = t & 3;
  int s = src[e], d = dst[e];
  const float4* kp = (const float4*)(krel + (size_t)s * 256 + h * 64);
  const float4* qp = (const float4*)(q + (size_t)d * 256 + h * 64);
  float sum = 0.f;
#pragma unroll
  for (int i = 0; i < 16; ++i) {
    float4 a = kp[i], b = qp[i];
    sum += a.x * b.x + a.y * b.y + a.z * b.z + a.w * b.w;
  }
  float al = sum * prior[h] * scale;
  alphabuf[t] = al;
  atomicMax(&amaxU[d * 4 + h], f32_ordered_key(al));
}

// Edge pass B: ex = exp(alpha - amax[dst,h]); den[dst,h] += ex (atomic).
__global__ void hgt_expnorm_kernel(const int* __restrict__ dst,
                                   const unsigned* __restrict__ amaxU,
                                   float* __restrict__ alphabuf,
                                   float* __restrict__ den, int E) {
  int t = blockIdx.x * blockDim.x + threadIdx.x;
  if (t >= E * 4) return;
  int e = t >> 2, h = t & 3;
  int d = dst[e];
  float amax = f32_ordered_unkey(amaxU[d * 4 + h]);
  float ex = expf(alphabuf[t] - amax);
  alphabuf[t] = ex;
  atomicAdd(&den[d * 4 + h], ex);
}

// Edge pass C: agg[dst, c] += m_rel[src, c] * ex/(den+eps). One block per edge,
// 256 threads => fully coalesced reads + coalesced f32 atomics.
__global__ void hgt_scatter_kernel(const float* __restrict__ mrel,
                                   const int* __restrict__ src,
                                   const int* __restrict__ dst,
                                   const float* __restrict__ exbuf,
                                   const float* __restrict__ den,
                                   float* __restrict__ agg, int E) {
  int e = blockIdx.x;
  if (e >= E) return;
  int c = threadIdx.x;          // 0..255
  int h = c >> 6;
  int s = src[e], d = dst[e];
  float w = exbuf[e * 4 + h] / (den[d * 4 + h] + 1e-16f);
  atomicAdd(&agg[(size_t)d * 256 + c], mrel[(size_t)s * 256 + c] * w);
}

// Exact GELU in place (jax.nn.gelu approximate=False).
__global__ void hgt_gelu_kernel(float* __restrict__ a, size_t n) {
  size_t t = (size_t)blockIdx.x * blockDim.x + threadIdx.x;
  if (t >= n) return;
  float x = a[t];
  a[t] = 0.5f * x * (1.0f + erff(x * 0.70710678118654752f));
}

// h = relu( sigmoid(skip)*o + (1-sigmoid(skip))*h )  in place.
__global__ void hgt_combine_relu(float* __restrict__ h,
                                 const float* __restrict__ o,
                                 const float* __restrict__ skip, size_t n) {
  size_t t = (size_t)blockIdx.x * blockDim.x + threadIdx.x;
  if (t >= n) return;
  float s = 1.0f / (1.0f + expf(-skip[0]));
  float v = s * o[t] + (1.0f - s) * h[t];
  h[t] = v > 0.0f ? v : 0.0f;
}

// ---------------------------------------------------------------------------
extern "C" void kernel_launch(void* const* d_in, const int* in_sizes, int n_in,
                              void* d_out, int out_size, void* d_ws, size_t ws_size,
                              hipStream_t stream) {
  const float* x_user = (const float*)d_in[0];
  const float* x_item = (const float*)d_in[1];
  // edge indices: row 0 = src (E entries), row 1 = dst (E entries)
  const int* ei[3] = {(const int*)d_in[2], (const int*)d_in[3], (const int*)d_in[4]};
  int Es[3] = {in_sizes[2] / 2, in_sizes[3] / 2, in_sizes[4] / 2};

  const int NU = in_sizes[0] / 128;   // 20000
  const int NI = in_sizes[1] / 64;    // 20000
  const int Ns[2] = {NU, NI};
  // edge-type order: 0 = user->clicks->item, 1 = item->clicked_by->user,
  //                  2 = user->follows->user
  const int srcT[3] = {0, 1, 0};
  const int dstT[3] = {1, 0, 0};

  // ---- parameters: jax tree_flatten order (dict keys sorted) -------------
  const float* P = (const float*)d_in[5];
  size_t off = 0;
  int leaf = 0;
  auto take = [&](size_t n) -> const float* {
    if (n_in > 6) return (const float*)d_in[5 + (leaf++)];  // leaves passed separately
    const float* p = P + off; off += n; return p;
  };
  // in_proj: item{b,w}, user{b,w}
  const float* ip_b[2]; const float* ip_w[2];
  ip_b[1] = take(256); ip_w[1] = take(64 * 256);
  ip_b[0] = take(256); ip_w[0] = take(128 * 256);
  struct LP {
    const float *a_b[2], *a_w[2], *k_b[2], *k_w[2], *q_b[2], *q_w[2];
    const float *rel_a[3], *rel_m[3], *rel_p[3];
    const float *skip[2];
    const float *v_b[2], *v_w[2];
  } L[2];
  for (int l = 0; l < 2; ++l) {
    LP& lp = L[l];
    // keys sorted: a, k, q, rel, skip, v ; node types sorted: item, user
    lp.a_b[1] = take(256); lp.a_w[1] = take(65536);
    lp.a_b[0] = take(256); lp.a_w[0] = take(65536);
    lp.k_b[1] = take(256); lp.k_w[1] = take(65536);
    lp.k_b[0] = take(256); lp.k_w[0] = take(65536);
    lp.q_b[1] = take(256); lp.q_w[1] = take(65536);
    lp.q_b[0] = take(256); lp.q_w[0] = take(65536);
    // rel keys sorted: item__clicked_by__user, user__clicks__item, user__follows__user
    const float* ra[3]; const float* rm[3]; const float* rp[3];
    for (int i = 0; i < 3; ++i) {
      ra[i] = take(4 * 64 * 64); rm[i] = take(4 * 64 * 64); rp[i] = take(4);
    }
    lp.rel_a[0] = ra[1]; lp.rel_m[0] = rm[1]; lp.rel_p[0] = rp[1]; // clicks
    lp.rel_a[1] = ra[0]; lp.rel_m[1] = rm[0]; lp.rel_p[1] = rp[0]; // clicked_by
    lp.rel_a[2] = ra[2]; lp.rel_m[2] = rm[2]; lp.rel_p[2] = rp[2]; // follows
    lp.skip[1] = take(1); lp.skip[0] = take(1);
    lp.v_b[1] = take(256); lp.v_w[1] = take(65536);
    lp.v_b[0] = take(256); lp.v_w[0] = take(65536);
  }
  const float* op_b[2]; const float* op_w[2];
  op_b[1] = take(128); op_w[1] = take(256 * 128);
  op_b[0] = take(128); op_w[0] = take(256 * 128);

  // ---- workspace layout ---------------------------------------------------
  float* wsf = (float*)d_ws;
  const size_t HS[2] = {(size_t)NU * 256, (size_t)NI * 256};
  float *h[2], *kb[2], *qb[2], *vb[2];
  h[0]  = wsf; wsf += HS[0]; h[1]  = wsf; wsf += HS[1];
  kb[0] = wsf; wsf += HS[0]; kb[1] = wsf; wsf += HS[1];
  qb[0] = wsf; wsf += HS[0]; qb[1] = wsf; wsf += HS[1];
  vb[0] = wsf; wsf += HS[0]; vb[1] = wsf; wsf += HS[1];
  float* relbuf = wsf; wsf += (HS[0] > HS[1] ? HS[0] : HS[1]);
  float* ebuf[3];
  for (int et = 0; et < 3; ++et) { ebuf[et] = wsf; wsf += (size_t)Es[et] * 4; }
  unsigned* amaxU[2]; float* den[2];
  amaxU[0] = (unsigned*)wsf; wsf += (size_t)NU * 4;
  amaxU[1] = (unsigned*)wsf; wsf += (size_t)NI * 4;
  den[0] = wsf; wsf += (size_t)NU * 4;
  den[1] = wsf; wsf += (size_t)NI * 4;
  // buffer re-use after attention passes (stream-sequential, safe):
  float* agg[2] = {kb[0], kb[1]};
  float* otmp = qb[0];

  auto gemm = [&](const float* A, const float* W, const float* b, float* C,
                  int M, int K, int N) {
    dim3 grid((M + 127) / 128, N / 64);
    hgt_gemm_bias<<<grid, dim3(128), 0, stream>>>(A, W, b, C, M, K, N);
  };
  auto head_gemm = [&](const float* A, const float* R, float* C, int M) {
    dim3 grid((M + 127) / 128, 4);
    hgt_head_gemm<<<grid, dim3(128), 0, stream>>>(A, R, C, M);
  };

  // ---- input projections --------------------------------------------------
  gemm(x_user, ip_w[0], ip_b[0], h[0], NU, 128, 256);
  gemm(x_item, ip_w[1], ip_b[1], h[1], NI, 64, 256);

  const float scale = 0.125f;  // 1/sqrt(64)

  for (int l = 0; l < 2; ++l) {
    LP& lp = L[l];
    // K/Q/V projections per node type
    for (int t = 0; t < 2; ++t) {
      gemm(h[t], lp.k_w[t], lp.k_b[t], kb[t], Ns[t], 256, 256);
      gemm(h[t], lp.q_w[t], lp.q_b[t], qb[t], Ns[t], 256, 256);
      gemm(h[t], lp.v_w[t], lp.v_b[t], vb[t], Ns[t], 256, 256);
    }
    // softmax state
    for (int t = 0; t < 2; ++t) {
      hipMemsetAsync(amaxU[t], 0, (size_t)Ns[t] * 4 * sizeof(unsigned), stream);
      hipMemsetAsync(den[t], 0, (size_t)Ns[t] * 4 * sizeof(float), stream);
    }
    // pass A: relation key transform + alpha + segment max
    for (int et = 0; et < 3; ++et) {
      head_gemm(kb[srcT[et]], lp.rel_a[et], relbuf, Ns[srcT[et]]);
      int nthr = Es[et] * 4;
      hgt_alpha_kernel<<<(nthr + 255) / 256, 256, 0, stream>>>(
          relbuf, qb[dstT[et]], ei[et], ei[et] + Es[et], lp.rel_p[et], scale,
          ebuf[et], amaxU[dstT[et]], Es[et]);
    }
    // pass B: exp + segment sum
    for (int et = 0; et < 3; ++et) {
      int nthr = Es[et] * 4;
      hgt_expnorm_kernel<<<(nthr + 255) / 256, 256, 0, stream>>>(
          ei[et] + Es[et], amaxU[dstT[et]], ebuf[et], den[dstT[et]], Es[et]);
    }
    // pass C: relation message transform + weighted scatter-add
    for (int t = 0; t < 2; ++t)
      hipMemsetAsync(agg[t], 0, HS[t] * sizeof(float), stream);  // k buffers free now
    for (int et = 0; et < 3; ++et) {
      head_gemm(vb[srcT[et]], lp.rel_m[et], relbuf, Ns[srcT[et]]);
      hgt_scatter_kernel<<<Es[et], 256, 0, stream>>>(
          relbuf, ei[et], ei[et] + Es[et], ebuf[et], den[dstT[et]],
          agg[dstT[et]], Es[et]);
    }
    // output transform + gated skip + relu
    for (int t = 0; t < 2; ++t) {
      hgt_gelu_kernel<<<(unsigned)((HS[t] + 255) / 256), 256, 0, stream>>>(agg[t], HS[t]);
      gemm(agg[t], lp.a_w[t], lp.a_b[t], otmp, Ns[t], 256, 256);
      hgt_combine_relu<<<(unsigned)((HS[t] + 255) / 256), 256, 0, stream>>>(
          h[t], otmp, lp.skip[t], HS[t]);
    }
  }

  // ---- output projections (user first, then item, per reference tuple) ----
  float* out = (float*)d_out;
  gemm(h[0], op_w[0], op_b[0], out, NU, 256, 128);
  gemm(h[1], op_w[1], op_b[1], out + (size_t)NU * 128, NI, 256, 128);
}